// TransRet_25503515804196
// MI455X (gfx1250) — compile-verified
//
#include <hip/hip_runtime.h>
#include <hip/hip_bf16.h>
#include <math.h>

#define D_MODEL 256
#define FFD     1024
#define TXTN    64
#define VIDN    2048
#define LTOT    2112
#define BATCH   16

typedef __attribute__((ext_vector_type(2))) float v2f;
typedef __attribute__((ext_vector_type(8))) float v8f;

// ---------------------------------------------------------------------------
// Full-precision f32 WMMA: D(16x16) = A(16x4) * B(4x16) + C
// ---------------------------------------------------------------------------
__device__ __forceinline__ v8f wmma4(v2f a, v2f b, v8f c) {
  return __builtin_amdgcn_wmma_f32_16x16x4_f32(
      /*neg_a=*/false, a, /*neg_b=*/false, b,
      /*c_mod=*/(short)0, c, /*reuse_a=*/false, /*reuse_b=*/false);
}

// A fragment: lane (m = lane&15) holds A[m][k0+2*hi], A[m][k0+2*hi+1]  (row-major A)
__device__ __forceinline__ v2f ldA(const float* A, int lda, int k0, int lane) {
  int m = lane & 15, k = k0 + 2 * (lane >> 4);
  const float* p = A + (size_t)m * lda + k;
  v2f r; r[0] = p[0]; r[1] = p[1]; return r;
}
// B fragment from row-major KxN matrix: B[k][n], B[k+1][n] (stride ldb apart)
__device__ __forceinline__ v2f ldB_rm(const float* Bm, int ldb, int k0, int n0, int lane) {
  int k = k0 + 2 * (lane >> 4), n = n0 + (lane & 15);
  const float* p = Bm + (size_t)k * ldb + n;
  v2f r; r[0] = p[0]; r[1] = p[ldb]; return r;
}
// B fragment where B[k][n] = M2[n][k]  (second operand transposed; contiguous in k)
__device__ __forceinline__ v2f ldB_tr(const float* M2, int ldm, int n0, int k0, int lane) {
  int n = n0 + (lane & 15), k = k0 + 2 * (lane >> 4);
  const float* p = M2 + (size_t)n * ldm + k;
  v2f r; r[0] = p[0]; r[1] = p[1]; return r;
}

// xPos rotary factors for column c, position p inside a segment of length L
// (minoff = L/2).  down=true -> downscale (K side).
__device__ __forceinline__ void xpos_cs(int p, int c, float minoff, bool down,
                                        float& cs, float& sn) {
  int i = c >> 1;
  float sv   = (2.f * (float)i + 102.4f) * (1.f / 358.4f);      // (2i + 0.4d)/(1.4d)
  float pw   = ((float)p - minoff) * (1.f / 512.f);             // (p + min_pos)/SCALE_BASE
  float sc   = powf(sv, pw);
  if (down) sc = 1.f / sc;
  float invf = exp2f(-(float)i * 0.10381025296f);               // 10000^(-i/128)
  float ang  = (float)p * invf;
  float s0, c0; sincosf(ang, &s0, &c0);
  cs = c0 * sc; sn = s0 * sc;
}

// ---------------------------------------------------------------------------
// K1: LayerNorm1 over concat(txt,vid) rows + txt-segment xPos Q/K and V
// ---------------------------------------------------------------------------
__global__ void ln1_txt_kernel(const float* __restrict__ txt, const float* __restrict__ vid,
                               const float* __restrict__ g, const float* __restrict__ be,
                               float* __restrict__ src2, float* __restrict__ Qb,
                               float* __restrict__ Kb, float* __restrict__ Vb) {
  int row = blockIdx.x;                       // 0 .. B*L-1
  int b = row / LTOT, l = row % LTOT;
  int c = threadIdx.x;
  float x = (l < TXTN) ? txt[((size_t)b * TXTN + l) * D_MODEL + c]
                       : vid[((size_t)b * VIDN + (l - TXTN)) * D_MODEL + c];
  float s = x, s2 = x * x;
  for (int off = 16; off > 0; off >>= 1) {
    s  += __shfl_down(s,  off, 32);
    s2 += __shfl_down(s2, off, 32);
  }
  __shared__ float red[16];
  __shared__ float mv[2];
  int wv = threadIdx.x >> 5, ln = threadIdx.x & 31;
  if (ln == 0) { red[wv] = s; red[wv + 8] = s2; }
  __syncthreads();
  if (threadIdx.x == 0) {
    float S = 0.f, S2 = 0.f;
    for (int i = 0; i < 8; i++) { S += red[i]; S2 += red[i + 8]; }
    float mu = S / 256.f;
    mv[0] = mu;
    mv[1] = rsqrtf(S2 / 256.f - mu * mu + 1e-5f);
  }
  __syncthreads();
  float y = (x - mv[0]) * mv[1] * g[c] + be[c];
  src2[(size_t)row * D_MODEL + c] = y;
  if (l < TXTN) {                              // txt: Q=xpos(y), K=xpos_down(y), V=y
    Vb[(size_t)row * D_MODEL + c] = y;
    float part = __shfl_xor(y, 1, 32);         // partner column (c^1)
    float cs, sn; xpos_cs(l, c, 32.f, false, cs, sn);
    float q = (c & 1) ? (y * cs + part * sn) : (y * cs - part * sn);
    Qb[(size_t)row * D_MODEL + c] = q;
    float cs2, sn2; xpos_cs(l, c, 32.f, true, cs2, sn2);
    float k = (c & 1) ? (y * cs2 + part * sn2) : (y * cs2 - part * sn2);
    Kb[(size_t)row * D_MODEL + c] = k;
  }
}

// ---------------------------------------------------------------------------
// K4/K6: generic LayerNorm (one 256-col row per 256-thread block)
// ---------------------------------------------------------------------------
__global__ void ln_kernel(const float* __restrict__ in, float* __restrict__ out,
                          const float* __restrict__ g, const float* __restrict__ be) {
  int row = blockIdx.x;
  int c = threadIdx.x;
  float x = in[(size_t)row * D_MODEL + c];
  float s = x, s2 = x * x;
  for (int off = 16; off > 0; off >>= 1) {
    s  += __shfl_down(s,  off, 32);
    s2 += __shfl_down(s2, off, 32);
  }
  __shared__ float red[16];
  __shared__ float mv[2];
  int wv = threadIdx.x >> 5, ln = threadIdx.x & 31;
  if (ln == 0) { red[wv] = s; red[wv + 8] = s2; }
  __syncthreads();
  if (threadIdx.x == 0) {
    float S = 0.f, S2 = 0.f;
    for (int i = 0; i < 8; i++) { S += red[i]; S2 += red[i + 8]; }
    float mu = S / 256.f;
    mv[0] = mu;
    mv[1] = rsqrtf(S2 / 256.f - mu * mu + 1e-5f);
  }
  __syncthreads();
  out[(size_t)row * D_MODEL + c] = (x - mv[0]) * mv[1] * g[c] + be[c];
}

// ---------------------------------------------------------------------------
// K2: vid-segment QKV GEMM (grid.z selects W_Q/W_K/W_V) with fused xPos.
// Block = 4 waves, computes one 16-row x 256-col tile.
// ---------------------------------------------------------------------------
__global__ void vid_qkv_kernel(const float* __restrict__ src2,
                               const float* __restrict__ WQ, const float* __restrict__ WK,
                               const float* __restrict__ WV,
                               float* __restrict__ Qb, float* __restrict__ Kb,
                               float* __restrict__ Vb) {
  int which = blockIdx.z;
  const float* W = (which == 0) ? WQ : (which == 1) ? WK : WV;
  float* Ob = (which == 0) ? Qb : (which == 1) ? Kb : Vb;
  int g0 = blockIdx.x * 16;                     // vid-row tile start (never straddles batch)
  int b = g0 / VIDN, p0 = g0 % VIDN;
  int lane = threadIdx.x & 31, wave = threadIdx.x >> 5;
  int lrow = lane & 15, lhi = lane >> 4;
  const float* Abase = src2 + ((size_t)(b * LTOT + TXTN + p0)) * D_MODEL;
  v8f acc[4];
  for (int t = 0; t < 4; t++) for (int r = 0; r < 8; r++) acc[t][r] = 0.f;
  int n0w = wave * 64;
  for (int k0 = 0; k0 < D_MODEL; k0 += 4) {
    v2f a = ldA(Abase, D_MODEL, k0, lane);
    for (int t = 0; t < 4; t++) {
      v2f bf = ldB_rm(W, D_MODEL, k0, n0w + 16 * t, lane);
      acc[t] = wmma4(a, bf, acc[t]);
    }
  }
  for (int t = 0; t < 4; t++) {
    int c = n0w + 16 * t + lrow;
    for (int r = 0; r < 8; r++) {
      int rr = r + 8 * lhi;
      int p = p0 + rr;
      float val  = acc[t][r];
      float part = __shfl_xor(val, 1, 32);      // column c^1, same row (C-fragment layout)
      float res;
      if (which == 2) {
        res = val;                              // V: no rotary
      } else {
        float cs, sn; xpos_cs(p, c, 1024.f, which == 1, cs, sn);
        res = (c & 1) ? (val * cs + part * sn) : (val * cs - part * sn);
      }
      Ob[((size_t)(b * LTOT + TXTN + p)) * D_MODEL + c] = res;
    }
  }
}

// ---------------------------------------------------------------------------
// K3: fused retention attention.  Block = 4 waves = one 16-row Q tile.
// Streams K/V in 16-row chunks; 16x16 score tile lives in LDS only.
// ---------------------------------------------------------------------------
__global__ void attn_kernel(const float* __restrict__ Qb, const float* __restrict__ Kb,
                            const float* __restrict__ Vb,
                            const float* __restrict__ txt, const float* __restrict__ vid,
                            float* __restrict__ x1) {
  int blk = blockIdx.x;
  int b = blk / (LTOT / 16), mt = blk % (LTOT / 16);
  int m0 = mt * 16;
  int lane = threadIdx.x & 31, wave = threadIdx.x >> 5;
  int lrow = lane & 15, lhi = lane >> 4;
  __shared__ float Spart[4][256];
  __shared__ float Sfin[16 * 18];               // stride 18 keeps float2 reads 8B-aligned
  const float* Qrow = Qb + ((size_t)(b * LTOT + m0)) * D_MODEL;
  v8f acc[4];
  for (int t = 0; t < 4; t++) for (int r = 0; r < 8; r++) acc[t][r] = 0.f;

  for (int n0 = 0; n0 < LTOT; n0 += 16) {
    // --- partial S = Q K^T over this wave's 64-wide k slice ---
    const float* Krow = Kb + ((size_t)(b * LTOT + n0)) * D_MODEL;
    v8f s; for (int r = 0; r < 8; r++) s[r] = 0.f;
    for (int kk = 0; kk < 64; kk += 4) {
      int k0 = wave * 64 + kk;
      v2f a  = ldA(Qrow, D_MODEL, k0, lane);
      v2f bf = ldB_tr(Krow, D_MODEL, 0, k0, lane);   // B[k][n] = K[n0+n][k]
      s = wmma4(a, bf, s);
    }
    for (int r = 0; r < 8; r++) Spart[wave][(r + 8 * lhi) * 16 + lrow] = s[r];
    __syncthreads();
    // --- wave0 reduces the 4 partials and applies the retention mask ---
    if (wave == 0) {
      for (int j = 0; j < 8; j++) {
        int idx = lane + 32 * j;
        float v = Spart[0][idx] + Spart[1][idx] + Spart[2][idx] + Spart[3][idx];
        int rr = idx >> 4, cc = idx & 15;
        int gi = m0 + rr, gj = n0 + cc;
        float mval = 1.f;
        if (gi >= TXTN && gj >= TXTN) {               // vid x vid block: gamma^(i-j) causal
          int d = gi - gj;
          mval = (d >= 0) ? exp2f(-0.15200309344505f * (float)d) : 0.f;
        }
        Sfin[rr * 18 + cc] = v * mval;
      }
    }
    __syncthreads();
    // --- acc += S * V, each wave owns a 64-col slice of the output ---
    const float* Vrow = Vb + ((size_t)(b * LTOT + n0)) * D_MODEL;
    for (int k0 = 0; k0 < 16; k0 += 4) {
      int kk2 = k0 + 2 * lhi;
      const float* ps = &Sfin[lrow * 18 + kk2];
      v2f a2; a2[0] = ps[0]; a2[1] = ps[1];
      for (int t = 0; t < 4; t++) {
        v2f bf = ldB_rm(Vrow, D_MODEL, k0, wave * 64 + 16 * t, lane);
        acc[t] = wmma4(a2, bf, acc[t]);
      }
    }
    __syncthreads();                                  // Sfin/Spart reuse next iteration
  }
  // --- epilogue: x1 = src + attn ---
  for (int t = 0; t < 4; t++) {
    int c = wave * 64 + 16 * t + lrow;
    for (int r = 0; r < 8; r++) {
      int l = m0 + r + 8 * lhi;
      float sv = (l < TXTN) ? txt[((size_t)b * TXTN + l) * D_MODEL + c]
                            : vid[((size_t)b * VIDN + (l - TXTN)) * D_MODEL + c];
      x1[((size_t)(b * LTOT + l)) * D_MODEL + c] = sv + acc[t][r];
    }
  }
}

// ---------------------------------------------------------------------------
// K5: fused FFN: h = gelu(ln2 @ W1^T + b1) staged in LDS, out = h @ W2^T + b2 + x1
// Block = 4 waves = one 16-row tile.  64 KB static LDS for h (16 x 1024 f32).
// ---------------------------------------------------------------------------
__global__ void ffn_kernel(const float* __restrict__ ln2, const float* __restrict__ w1,
                           const float* __restrict__ b1, const float* __restrict__ w2,
                           const float* __restrict__ b2, const float* __restrict__ x1,
                           float* __restrict__ x2) {
  int m0g = blockIdx.x * 16;                    // global row (b*L+l, rows independent)
  int lane = threadIdx.x & 31, wave = threadIdx.x >> 5;
  int lrow = lane & 15, lhi = lane >> 4;
  __shared__ float hb[16 * 1024];
  const float* Abase = ln2 + (size_t)m0g * D_MODEL;
  // phase A: h strips (each wave: 4 strips of 16x64 across its 256 cols)
  for (int st = 0; st < 4; st++) {
    int nb = wave * 256 + st * 64;
    v8f acc[4];
    for (int t = 0; t < 4; t++) for (int r = 0; r < 8; r++) acc[t][r] = 0.f;
    for (int k0 = 0; k0 < D_MODEL; k0 += 4) {
      v2f a = ldA(Abase, D_MODEL, k0, lane);
      for (int t = 0; t < 4; t++) {
        v2f bf = ldB_tr(w1, D_MODEL, nb + 16 * t, k0, lane);  // B[k][n]=w1[n][k]
        acc[t] = wmma4(a, bf, acc[t]);
      }
    }
    for (int t = 0; t < 4; t++) {
      int n = nb + 16 * t + lrow;
      for (int r = 0; r < 8; r++) {
        int rr = r + 8 * lhi;
        float v = acc[t][r] + b1[n];
        hb[rr * 1024 + n] = 0.5f * v * (1.f + erff(v * 0.70710678118655f)); // exact gelu
      }
    }
  }
  __syncthreads();
  // phase B: out = h @ w2^T, A fragments from LDS
  v8f acc[4];
  for (int t = 0; t < 4; t++) for (int r = 0; r < 8; r++) acc[t][r] = 0.f;
  for (int k0 = 0; k0 < FFD; k0 += 4) {
    int kk = k0 + 2 * lhi;
    const float* ps = &hb[lrow * 1024 + kk];
    v2f a2; a2[0] = ps[0]; a2[1] = ps[1];
    for (int t = 0; t < 4; t++) {
      v2f bf = ldB_tr(w2, FFD, wave * 64 + 16 * t, k0, lane); // B[k][n]=w2[n][k]
      acc[t] = wmma4(a2, bf, acc[t]);
    }
  }
  for (int t = 0; t < 4; t++) {
    int c = wave * 64 + 16 * t + lrow;
    for (int r = 0; r < 8; r++) {
      int rr = r + 8 * lhi;
      size_t off = (size_t)(m0g + rr) * D_MODEL + c;
      x2[off] = acc[t][r] + b2[c] + x1[off];
    }
  }
}

// ---------------------------------------------------------------------------
extern "C" void kernel_launch(void* const* d_in, const int* in_sizes, int n_in,
                              void* d_out, int out_size, void* d_ws, size_t ws_size,
                              hipStream_t stream) {
  const float* txt = (const float*)d_in[0];
  const float* vid = (const float*)d_in[1];
  const float* WQ  = (const float*)d_in[2];
  const float* WK  = (const float*)d_in[3];
  const float* WV  = (const float*)d_in[4];
  const float* w1  = (const float*)d_in[5];
  const float* b1  = (const float*)d_in[6];
  const float* w2  = (const float*)d_in[7];
  const float* b2  = (const float*)d_in[8];
  const float* g1  = (const float*)d_in[9];
  const float* be1 = (const float*)d_in[10];
  const float* g2  = (const float*)d_in[11];
  const float* be2 = (const float*)d_in[12];
  const float* g3  = (const float*)d_in[13];
  const float* be3 = (const float*)d_in[14];

  float* ws = (float*)d_ws;
  const size_t NT = (size_t)BATCH * LTOT * D_MODEL;   // 8,650,752 floats = 34.6 MB
  float* src2 = ws;                                    // LN1 out -> later reused as LN2 out
  float* Qb   = ws + NT;                               // later reused as x2
  float* Kb   = ws + 2 * NT;
  float* Vb   = ws + 3 * NT;
  float* x1   = ws + 4 * NT;                           // src + attn
  float* ln2o = src2;
  float* x2   = Qb;
  float* outp = (float*)d_out;

  int nrows = BATCH * LTOT;                            // 33792
  ln1_txt_kernel<<<nrows, 256, 0, stream>>>(txt, vid, g1, be1, src2, Qb, Kb, Vb);
  vid_qkv_kernel<<<dim3(BATCH * VIDN / 16, 1, 3), 128, 0, stream>>>(src2, WQ, WK, WV, Qb, Kb, Vb);
  attn_kernel<<<nrows / 16, 128, 0, stream>>>(Qb, Kb, Vb, txt, vid, x1);
  ln_kernel<<<nrows, 256, 0, stream>>>(x1, ln2o, g2, be2);
  ffn_kernel<<<nrows / 16, 128, 0, stream>>>(ln2o, w1, b1, w2, b2, x1, x2);
  ln_kernel<<<nrows, 256, 0, stream>>>(x2, outp, g3, be3);
}